// CRF_7370163880062
// MI455X (gfx1250) — compile-verified
//
#include <hip/hip_runtime.h>
#include <math.h>

// CRF forward (partition function) for NUM_TAGS=128, SEQ_LEN=8192.
// Linear-domain reformulation: z = log( e_START^T · Π_t (E·diag(g_t)) · E[:,END]-weights )
// with E = exp(trans), g_t = exp(emit_row_t), computed as a chunked chain of
// 128x128x128 fp32 WMMA matmuls with per-step max rescaling.

typedef __attribute__((ext_vector_type(2))) float v2f;
typedef __attribute__((ext_vector_type(8))) float v8f;

#define NTAG   128
#define PITCH  132                 // LDS row pitch (floats) to avoid bank conflicts
#define MSZ    (NTAG * PITCH)      // pitched 128x128 matrix in LDS
#define ESZ    (NTAG * NTAG)       // 16384 floats, B-fragment swizzled
#define CHUNK  32
#define START_TAG 0
#define END_TAG   1

// wave32 max-reduce then atomic max into a shared int holding float bits.
// All values are >= 0, so integer compare == float compare on the bit patterns.
__device__ __forceinline__ void wave_max_atomic(float v, int* slot) {
    for (int off = 16; off > 0; off >>= 1)
        v = fmaxf(v, __shfl_xor(v, off, 32));
    if ((threadIdx.x & 31) == 0)
        atomicMax(slot, __float_as_int(v));
}

// One 128x128x128 step: D = (Mc * invm) x Bsw, column-scaled by g (or 1.0),
// stored pitched into Mn; returns after contributing to the shared max slot.
// Bsw layout: element (k,n) of the 128x128 B matrix at
//   idx = (k>>2)*512 + ((k>>1)&1)*256 + n*2 + (k&1)
// i.e. per 4-row K-block, [half][n][v] so each lane's 2 B values are a b64.
__device__ __forceinline__ void matmul_step(const float* __restrict__ Mc,
                                            const float* __restrict__ Bsw,
                                            float* __restrict__ Mn,
                                            const float* __restrict__ sG, // may be nullptr
                                            float invm, int* maxSlot) {
    const int lane = threadIdx.x & 31;
    const int wave = threadIdx.x >> 5;          // 8 waves: wave = output row-tile
    const int l16  = lane & 15;
    const int half = lane >> 4;

    v8f acc[8];
    #pragma unroll
    for (int jc = 0; jc < 8; ++jc) acc[jc] = (v8f)0.0f;

    const int rowoff = (wave * 16 + l16) * PITCH;
    for (int kk = 0; kk < 32; ++kk) {
        v2f a = *(const v2f*)(Mc + rowoff + kk * 4 + 2 * half);
        a *= invm;
        const float* eb = Bsw + kk * 512 + half * 256 + l16 * 2;
        #pragma unroll
        for (int jc = 0; jc < 8; ++jc) {
            v2f b = *(const v2f*)(eb + jc * 32);
            acc[jc] = __builtin_amdgcn_wmma_f32_16x16x4_f32(
                false, a, false, b, (short)0, acc[jc], false, false);
        }
    }

    float lmax = 0.0f;
    #pragma unroll
    for (int jc = 0; jc < 8; ++jc) {
        const float gv = sG ? sG[jc * 16 + l16] : 1.0f;
        #pragma unroll
        for (int v = 0; v < 8; ++v) {
            float val = acc[jc][v] * gv;
            lmax = fmaxf(lmax, val);
            Mn[(wave * 16 + 8 * half + v) * PITCH + jc * 16 + l16] = val;
        }
    }
    wave_max_atomic(lmax, maxSlot);
}

// ---------------- Phase A: per-chunk transfer-matrix products -----------------
__global__ __launch_bounds__(256) void crf_chunk_kernel(
        const int* __restrict__ x, const float* __restrict__ emit,
        const float* __restrict__ trans, float* __restrict__ outMats,
        float* __restrict__ outScales, int chunkLen, int totalT) {
    extern __shared__ float dyn[];
    float* Esw = dyn;               // ESZ floats, swizzled B layout of exp(trans)
    float* M0  = dyn + ESZ;         // MSZ floats (pitched)
    float* M1  = dyn + ESZ + MSZ;   // MSZ floats (pitched)
    __shared__ float sG[NTAG];
    __shared__ int   sMaxBits[2];
    __shared__ float sLog;

    const int tid = threadIdx.x;
    const int chunk = blockIdx.x;
    const int t0 = chunk * chunkLen;
    const int L  = min(chunkLen, totalT - t0);
    if (L <= 0) return;

    // Build E in swizzled B-fragment order: idx -> (kk,h,n,v)
    for (int idx = tid; idx < ESZ; idx += 256) {
        const int v = idx & 1, n = (idx >> 1) & 127, h = (idx >> 8) & 1, kk = idx >> 9;
        const int r = kk * 4 + 2 * h + v;
        Esw[idx] = expf(trans[r * NTAG + n]);
    }
    if (tid == 0) { sLog = 0.0f; sMaxBits[0] = 0; sMaxBits[1] = 0; }
    __syncthreads();

    float* Mc = M0;
    float* Mn = M1;

    // step 0: M = E * diag(g_0)  (raw, max -> slot 0)
    {
        const int tok = x[t0 + 1];
        if (tid < NTAG) sG[tid] = expf(emit[(size_t)tok * NTAG + tid]);
        __syncthreads();
        float lmax = 0.0f;
        for (int idx = tid; idx < ESZ; idx += 256) {
            const int rr = idx >> 7, cc = idx & 127;
            const float val = expf(trans[idx]) * sG[cc];
            Mn[rr * PITCH + cc] = val;
            lmax = fmaxf(lmax, val);
        }
        wave_max_atomic(lmax, &sMaxBits[0]);
        __syncthreads();
        float* t = Mc; Mc = Mn; Mn = t;
    }

    // steps 1..L-1: M <- (M/m_prev) * E * diag(g_s); sLog += log(m_prev)
    for (int s = 1; s < L; ++s) {
        const int p = s & 1;
        const float mprev = __int_as_float(sMaxBits[p ^ 1]);
        const float invm  = 1.0f / mprev;
        if (tid == 0) { sLog += logf(mprev); sMaxBits[p] = 0; }
        const int tok = x[t0 + s + 1];
        if (tid < NTAG) sG[tid] = expf(emit[(size_t)tok * NTAG + tid]);
        __syncthreads();

        matmul_step(Mc, Esw, Mn, sG, invm, &sMaxBits[p]);
        __syncthreads();
        float* t = Mc; Mc = Mn; Mn = t;
    }

    // Emit raw final matrix + scale (true matrix = raw * exp(sLog)).
    float* outp = outMats + (size_t)chunk * ESZ;
    for (int idx = tid; idx < ESZ; idx += 256)
        outp[idx] = Mc[(idx >> 7) * PITCH + (idx & 127)];
    if (tid == 0) outScales[chunk] = sLog;
}

// ---------------- Phase B: tree-combine groups of 8 matrices ------------------
__global__ __launch_bounds__(256) void crf_combine_kernel(
        const float* __restrict__ inMats, const float* __restrict__ inScales,
        float* __restrict__ outMats, float* __restrict__ outScales,
        int groupSize, int nIn) {
    extern __shared__ float dyn[];
    float* Bsw = dyn;               // ESZ, swizzled right operand
    float* R0  = dyn + ESZ;         // MSZ pitched running product
    float* R1  = dyn + ESZ + MSZ;
    __shared__ int   sMaxBits[2];
    __shared__ float sLog;

    const int tid = threadIdx.x;
    const int base = blockIdx.x * groupSize;
    const int cnt = min(groupSize, nIn - base);
    if (cnt <= 0) return;

    if (tid == 0) { sMaxBits[0] = 0; sMaxBits[1] = 0; sLog = inScales[base]; }
    __syncthreads();

    // init R = inMats[base] (raw), max -> slot 0
    {
        const float* src = inMats + (size_t)base * ESZ;
        float lmax = 0.0f;
        for (int idx = tid; idx < ESZ; idx += 256) {
            const float v = src[idx];
            R0[(idx >> 7) * PITCH + (idx & 127)] = v;
            lmax = fmaxf(lmax, v);
        }
        wave_max_atomic(lmax, &sMaxBits[0]);
        __syncthreads();
    }

    float* Mc = R0;
    float* Mn = R1;
    for (int k = 1; k < cnt; ++k) {
        const int p = k & 1;
        const float mprev = __int_as_float(sMaxBits[p ^ 1]);
        const float invm  = 1.0f / mprev;
        if (tid == 0) { sLog += logf(mprev) + inScales[base + k]; sMaxBits[p] = 0; }
        const float* rm = inMats + (size_t)(base + k) * ESZ;
        for (int idx = tid; idx < ESZ; idx += 256) {
            const int v = idx & 1, n = (idx >> 1) & 127, h = (idx >> 8) & 1, kk = idx >> 9;
            Bsw[idx] = rm[(kk * 4 + 2 * h + v) * NTAG + n];
        }
        __syncthreads();

        matmul_step(Mc, Bsw, Mn, nullptr, invm, &sMaxBits[p]);
        __syncthreads();
        float* t = Mc; Mc = Mn; Mn = t;
    }

    float* outp = outMats + (size_t)blockIdx.x * ESZ;
    for (int idx = tid; idx < ESZ; idx += 256)
        outp[idx] = Mc[(idx >> 7) * PITCH + (idx & 127)];
    if (tid == 0) outScales[blockIdx.x] = sLog;
}

// ---------------- Phase C: vector sweep through the last few matrices ---------
__global__ __launch_bounds__(128) void crf_final_kernel(
        const float* __restrict__ mats, const float* __restrict__ scales, int n,
        const float* __restrict__ trans, float* __restrict__ out) {
    __shared__ float sA[NTAG];
    __shared__ float sR[NTAG];
    const int j = threadIdx.x;

    float a = mats[START_TAG * NTAG + j];   // e_START^T * P0 (raw)
    float logacc = scales[0];

    for (int c = 1; c < n; ++c) {
        sA[j] = a; sR[j] = a;
        __syncthreads();
        for (int off = 64; off > 0; off >>= 1) {
            if (j < off) sR[j] = fmaxf(sR[j], sR[j + off]);
            __syncthreads();
        }
        const float m = sR[0];
        const float* P = mats + (size_t)c * ESZ;
        float s = 0.0f;
        for (int i = 0; i < NTAG; ++i) s += sA[i] * P[i * NTAG + j];
        a = s / m;
        logacc += logf(m) + scales[c];
        __syncthreads();
    }

    // z = log( sum_j a_true[j] * exp(trans[j, END]) )
    sR[j] = a;
    __syncthreads();
    for (int off = 64; off > 0; off >>= 1) {
        if (j < off) sR[j] = fmaxf(sR[j], sR[j + off]);
        __syncthreads();
    }
    const float m = sR[0];
    __syncthreads();
    sR[j] = (a / m) * expf(trans[j * NTAG + END_TAG]);
    __syncthreads();
    for (int off = 64; off > 0; off >>= 1) {
        if (j < off) sR[j] += sR[j + off];
        __syncthreads();
    }
    if (j == 0) out[0] = logf(sR[0]) + logf(m) + logacc;
}

extern "C" void kernel_launch(void* const* d_in, const int* in_sizes, int n_in,
                              void* d_out, int out_size, void* d_ws, size_t ws_size,
                              hipStream_t stream) {
    const int*   x     = (const int*)d_in[0];
    const float* emit  = (const float*)d_in[1];
    const float* trans = (const float*)d_in[2];
    float* out = (float*)d_out;

    const int totalT  = in_sizes[0] - 1;                 // 8191 emit steps
    const int nChunks = (totalT + CHUNK - 1) / CHUNK;    // 256
    const int n1 = (nChunks + 7) / 8;                    // 32
    const int n2 = (n1 + 7) / 8;                         // 4

    float* bufA = (float*)d_ws;
    float* scA  = bufA + (size_t)nChunks * ESZ;
    float* bufB = scA + nChunks;
    float* scB  = bufB + (size_t)n1 * ESZ;
    float* bufC = scB + n1;
    float* scC  = bufC + (size_t)n2 * ESZ;

    const size_t dynBytes = (size_t)(ESZ + 2 * MSZ) * sizeof(float); // ~196 KB
    hipFuncSetAttribute(reinterpret_cast<const void*>(crf_chunk_kernel),
                        hipFuncAttributeMaxDynamicSharedMemorySize, (int)dynBytes);
    hipFuncSetAttribute(reinterpret_cast<const void*>(crf_combine_kernel),
                        hipFuncAttributeMaxDynamicSharedMemorySize, (int)dynBytes);

    crf_chunk_kernel<<<nChunks, 256, dynBytes, stream>>>(x, emit, trans, bufA, scA,
                                                         CHUNK, totalT);
    crf_combine_kernel<<<n1, 256, dynBytes, stream>>>(bufA, scA, bufB, scB, 8, nChunks);
    crf_combine_kernel<<<n2, 256, dynBytes, stream>>>(bufB, scB, bufC, scC, 8, n1);
    crf_final_kernel<<<1, 128, 0, stream>>>(bufC, scC, n2, trans, out);
}